// Network_7258494730380
// MI455X (gfx1250) — compile-verified
//
#include <hip/hip_runtime.h>

typedef __attribute__((ext_vector_type(2))) float v2f;
typedef __attribute__((ext_vector_type(8))) float v8f;
typedef __attribute__((ext_vector_type(4))) int   v4i;

#define EPSC 1e-12f

#if defined(__has_builtin)
#if __has_builtin(__builtin_amdgcn_global_load_async_to_lds_b128) && \
    __has_builtin(__builtin_amdgcn_s_wait_asynccnt)
#define HAVE_ASYNC_LDS 1
#endif
#endif

#ifdef HAVE_ASYNC_LDS
#define AS1 __attribute__((address_space(1)))
#define AS3 __attribute__((address_space(3)))
#endif

__device__ __forceinline__ float maxabs2(float a, float b) {
  float fa = __builtin_fabsf(a), fb = __builtin_fabsf(b);
  if (fa > fb) return a;
  if (fb > fa) return b;
  return fmaxf(a, b);   // tie -> positive one, matches reference (pos wins)
}

__device__ __forceinline__ float scs_act(float num, float xnorm, float ex) {
  float y = num / xnorm;
  float m = exp2f(ex * log2f(__builtin_fabsf(y) + EPSC));
  return copysignf(m, y);
}

// ---------------------------------------------------------------------------
// Kernel 1: scs conv1 (5x5, C=3 -> O=80, 32x32 -> 28x28) fused with 2x2
// max-abs pool -> (N,80,14,14).  Implicit GEMM, one block (8 waves) / image.
// M ordering permuted so 4 consecutive M = one 2x2 pool window (pool is
// intra-lane over accumulator VGPRs).  B fragments register-cached per Ntile.
// ---------------------------------------------------------------------------
__global__ __launch_bounds__(256)
void k_scs1(const float* __restrict__ x, const float* __restrict__ w1,
            const float* __restrict__ p1, const float* __restrict__ q1,
            float* __restrict__ t1p) {
  __shared__ __align__(16) float xs[3 * 32 * 32];  // image slab
  __shared__ __align__(16) float wnsp[38 * 80 * 2]; // wn pair-major [kp][o][2]
  __shared__ float xn[784];                         // patch norm, permuted M
  __shared__ float expo[80];
  __shared__ float wnorm[80];
  __shared__ int   koff[76];        // k -> lds offset (c*1024 + kh*32 + kw)

  const int tid = threadIdx.x;
  const int n = blockIdx.x;
  const float* xg = x + (size_t)n * 3072;

  float qv = q1[0] * 0.01f; qv *= qv;

#ifdef HAVE_ASYNC_LDS
  for (int i = tid * 4; i < 3072; i += 256 * 4)
    __builtin_amdgcn_global_load_async_to_lds_b128(
        (AS1 v4i*)(xg + i), (AS3 v4i*)&xs[i], 0, 0);
#else
  for (int i = tid; i < 3072; i += 256) xs[i] = xg[i];
#endif
  for (int k = tid; k < 76; k += 256) {
    int kk = (k < 75) ? k : 0;
    int c = kk / 25, r = kk % 25;
    koff[k] = c * 1024 + (r / 5) * 32 + (r % 5);
  }
  for (int o = tid; o < 80; o += 256) {
    float e = p1[o] * 0.1f; expo[o] = e * e;
    float s = 0.f;
    const float* w = w1 + o * 75;
    for (int k = 0; k < 75; k++) s += w[k] * w[k];
    wnorm[o] = __builtin_sqrtf(s + EPSC) + qv;
  }
#ifdef HAVE_ASYNC_LDS
  __builtin_amdgcn_s_wait_asynccnt(0);
#endif
  __syncthreads();
  // normalized weights, pair-major: wnsp[(kp*80+o)*2+j] = wn[2*kp+j][o]
  for (int i = tid; i < 38 * 80 * 2; i += 256) {
    int j = i & 1, t = i >> 1;
    int o = t % 80, kp = t / 80;
    int k = 2 * kp + j;
    wnsp[i] = (k < 75) ? w1[o * 75 + k] / wnorm[o] : 0.f;
  }
  for (int m = tid; m < 784; m += 256) {
    int pp = m >> 2, e = m & 3;
    int base = (2 * (pp / 14) + (e >> 1)) * 32 + 2 * (pp % 14) + (e & 1);
    float s = 0.f;
    for (int k = 0; k < 75; k++) { float v = xs[base + koff[k]]; s += v * v; }
    xn[m] = __builtin_sqrtf(s + EPSC) + qv;
  }
  __syncthreads();

  const int wave = tid >> 5, lane = tid & 31;
  const int lm = lane & 15, lh = lane >> 4;

  for (int nt = 0; nt < 5; nt++) {
    v2f breg[19];                       // B register cache for this Ntile
#pragma unroll
    for (int ks = 0; ks < 19; ks++) {
      int kp = ks * 2 + lh;             // = k0/2 with k0 = 4*ks + 2*lh
      breg[ks] = *(const v2f*)&wnsp[(kp * 80 + nt * 16 + lm) * 2];
    }
    for (int mt = wave; mt < 49; mt += 8) {
      int M = mt * 16 + lm;
      int pp = M >> 2, e = M & 3;
      int abase = (2 * (pp / 14) + (e >> 1)) * 32 + 2 * (pp % 14) + (e & 1);
      v8f acc = {0.f, 0.f, 0.f, 0.f, 0.f, 0.f, 0.f, 0.f};
#pragma unroll
      for (int ks = 0; ks < 19; ks++) {
        int k0 = ks * 4 + lh * 2;
        v2f a;
        a.x = xs[abase + koff[k0]];
        a.y = xs[abase + koff[k0 + 1]];
        acc = __builtin_amdgcn_wmma_f32_16x16x4_f32(false, a, false, breg[ks],
                                                    (short)0, acc, false, false);
      }
      int o = nt * 16 + lm;
      float ex = expo[o];
      int mbase = mt * 16 + 8 * lh;
      float v0 = scs_act(acc[0], xn[mbase + 0], ex);
      float v1 = scs_act(acc[1], xn[mbase + 1], ex);
      float v2 = scs_act(acc[2], xn[mbase + 2], ex);
      float v3 = scs_act(acc[3], xn[mbase + 3], ex);
      float v4 = scs_act(acc[4], xn[mbase + 4], ex);
      float v5 = scs_act(acc[5], xn[mbase + 5], ex);
      float v6 = scs_act(acc[6], xn[mbase + 6], ex);
      float v7 = scs_act(acc[7], xn[mbase + 7], ex);
      v2f r;
      r.x = maxabs2(maxabs2(v0, v1), maxabs2(v2, v3));
      r.y = maxabs2(maxabs2(v4, v5), maxabs2(v6, v7));
      int pp0 = 4 * mt + 2 * lh;
      *(v2f*)(t1p + ((size_t)n * 80 + o) * 196 + pp0) = r;   // 8B aligned
    }
  }
}

// ---------------------------------------------------------------------------
// Kernel 2: depthwise scs 3x3 (groups=80): (N,80,14,14) -> (N,80,12,12).
// No cross-channel reduction -> VALU kernel (0.2 GFLOP, negligible).
// ---------------------------------------------------------------------------
__global__ __launch_bounds__(256)
void k_dw(const float* __restrict__ in, const float* __restrict__ w,
          const float* __restrict__ p, const float* __restrict__ q,
          float* __restrict__ out) {
  int idx = blockIdx.x * 256 + threadIdx.x;
  if (idx >= 1024 * 80 * 144) return;
  int ow = idx % 12; int t = idx / 12;
  int oh = t % 12; t /= 12;
  int o = t % 80; int n = t / 80;
  float qv = q[0] * 0.01f; qv *= qv;
  const float* wc = w + o * 9;
  float ws2 = 0.f;
#pragma unroll
  for (int k = 0; k < 9; k++) ws2 += wc[k] * wc[k];
  float rwn = 1.f / (__builtin_sqrtf(ws2 + EPSC) + qv);
  const float* xc = in + ((size_t)n * 80 + o) * 196;
  float num = 0.f, ss = 0.f;
#pragma unroll
  for (int kh = 0; kh < 3; kh++)
#pragma unroll
    for (int kw = 0; kw < 3; kw++) {
      float v = xc[(oh + kh) * 14 + ow + kw];
      num += v * wc[kh * 3 + kw];
      ss += v * v;
    }
  float ee = p[o] * 0.1f; ee *= ee;
  out[idx] = scs_act(num * rwn, __builtin_sqrtf(ss + EPSC) + qv, ee);
}

// ---------------------------------------------------------------------------
// Kernel 3: pointwise scs (1x1, 80->40) as WMMA GEMM, fused with 2x2 max-abs
// pool: (N,80,12,12) -> (N,40,6,6).  M=144/image (pool-permuted), K=80,
// N=48pad.  B fragments register-cached per Ntile.
// ---------------------------------------------------------------------------
__global__ __launch_bounds__(256)
void k_pw(const float* __restrict__ in, const float* __restrict__ w2p,
          const float* __restrict__ p2p, const float* __restrict__ q2p,
          float* __restrict__ out) {
  __shared__ __align__(16) float xt[80 * 144];      // [k][m] transposed+permuted
  __shared__ __align__(16) float wnsp[40 * 48 * 2]; // wn pair-major
  __shared__ float xn[144];
  __shared__ float expo[48];
  __shared__ float wnorm[48];

  const int tid = threadIdx.x;
  const int n = blockIdx.x;
  float qv = q2p[0] * 0.01f; qv *= qv;

  const float* xg = in + (size_t)n * 80 * 144;
  for (int i = tid; i < 80 * 144; i += 256) {
    int k = i / 144, m = i % 144;
    int pp = m >> 2, e = m & 3;
    int oh = 2 * (pp / 6) + (e >> 1), ow = 2 * (pp % 6) + (e & 1);
    xt[i] = xg[k * 144 + oh * 12 + ow];
  }
  for (int o = tid; o < 48; o += 256) {
    if (o < 40) {
      float ee = p2p[o] * 0.1f; expo[o] = ee * ee;
      float s = 0.f;
      for (int k = 0; k < 80; k++) { float w = w2p[o * 80 + k]; s += w * w; }
      wnorm[o] = __builtin_sqrtf(s + EPSC) + qv;
    } else { expo[o] = 1.f; wnorm[o] = 1.f; }
  }
  __syncthreads();
  for (int i = tid; i < 40 * 48 * 2; i += 256) {
    int j = i & 1, t = i >> 1;
    int o = t % 48, kp = t / 48;
    int k = 2 * kp + j;
    wnsp[i] = (o < 40) ? w2p[o * 80 + k] / wnorm[o] : 0.f;
  }
  for (int m = tid; m < 144; m += 256) {
    float s = 0.f;
    for (int k = 0; k < 80; k++) { float v = xt[k * 144 + m]; s += v * v; }
    xn[m] = __builtin_sqrtf(s + EPSC) + qv;
  }
  __syncthreads();

  const int wave = tid >> 5, lane = tid & 31;
  const int lm = lane & 15, lh = lane >> 4;

  for (int nt = 0; nt < 3; nt++) {
    v2f breg[20];
#pragma unroll
    for (int ks = 0; ks < 20; ks++) {
      int kp = ks * 2 + lh;
      breg[ks] = *(const v2f*)&wnsp[(kp * 48 + nt * 16 + lm) * 2];
    }
    for (int mt = wave; mt < 9; mt += 8) {
      int M = mt * 16 + lm;
      v8f acc = {0.f, 0.f, 0.f, 0.f, 0.f, 0.f, 0.f, 0.f};
#pragma unroll
      for (int ks = 0; ks < 20; ks++) {
        int k0 = ks * 4 + lh * 2;
        v2f a;
        a.x = xt[k0 * 144 + M];
        a.y = xt[(k0 + 1) * 144 + M];
        acc = __builtin_amdgcn_wmma_f32_16x16x4_f32(false, a, false, breg[ks],
                                                    (short)0, acc, false, false);
      }
      int o = nt * 16 + lm;
      if (o < 40) {
        float ex = expo[o];
        int mbase = mt * 16 + 8 * lh;
        float v0 = scs_act(acc[0], xn[mbase + 0], ex);
        float v1 = scs_act(acc[1], xn[mbase + 1], ex);
        float v2 = scs_act(acc[2], xn[mbase + 2], ex);
        float v3 = scs_act(acc[3], xn[mbase + 3], ex);
        float v4 = scs_act(acc[4], xn[mbase + 4], ex);
        float v5 = scs_act(acc[5], xn[mbase + 5], ex);
        float v6 = scs_act(acc[6], xn[mbase + 6], ex);
        float v7 = scs_act(acc[7], xn[mbase + 7], ex);
        v2f r;
        r.x = maxabs2(maxabs2(v0, v1), maxabs2(v2, v3));
        r.y = maxabs2(maxabs2(v4, v5), maxabs2(v6, v7));
        int pp0 = 4 * mt + 2 * lh;
        *(v2f*)(out + ((size_t)n * 40 + o) * 36 + pp0) = r;  // 8B aligned
      }
    }
  }
}

// ---------------------------------------------------------------------------
// Kernel 4: scs conv3 (3x3, 40->40, 6x6 -> 4x4) + 4x4 max-abs pool + final
// linear (40->10), all fused.  Per image: M=16 (one tile), K=360, N=48(pad).
// One wave per image, 8 images per block.  Writes d_out directly.
// ---------------------------------------------------------------------------
__global__ __launch_bounds__(256)
void k_c3(const float* __restrict__ in, const float* __restrict__ w3,
          const float* __restrict__ p3, const float* __restrict__ q3,
          const float* __restrict__ wo, const float* __restrict__ bo,
          float* __restrict__ out) {
  __shared__ __align__(16) float wnsp[180 * 48 * 2]; // 69 KB, pair-major
  __shared__ __align__(16) float xs[8][1440];        // 46 KB (8 imgs (40,6,6))
  __shared__ float xn[8][16];
  __shared__ float po[8][48];
  __shared__ float expo[48];
  __shared__ float wnorm[48];
  __shared__ int   koff[360];

  const int tid = threadIdx.x;
  const int n0 = blockIdx.x * 8;
  float qv = q3[0] * 0.01f; qv *= qv;

#ifdef HAVE_ASYNC_LDS
  for (int i = tid * 4; i < 8 * 1440; i += 256 * 4)
    __builtin_amdgcn_global_load_async_to_lds_b128(
        (AS1 v4i*)(in + (size_t)n0 * 1440 + i), (AS3 v4i*)&xs[0][i], 0, 0);
#else
  for (int i = tid; i < 8 * 1440; i += 256)
    xs[i / 1440][i % 1440] = in[(size_t)n0 * 1440 + i];
#endif
  for (int k = tid; k < 360; k += 256) {
    int c = k / 9, r = k % 9;
    koff[k] = c * 36 + (r / 3) * 6 + (r % 3);
  }
  for (int o = tid; o < 48; o += 256) {
    if (o < 40) {
      float ee = p3[o] * 0.1f; expo[o] = ee * ee;
      float s = 0.f;
      for (int k = 0; k < 360; k++) { float w = w3[o * 360 + k]; s += w * w; }
      wnorm[o] = __builtin_sqrtf(s + EPSC) + qv;
    } else { expo[o] = 1.f; wnorm[o] = 1.f; }
  }
#ifdef HAVE_ASYNC_LDS
  __builtin_amdgcn_s_wait_asynccnt(0);
#endif
  __syncthreads();
  for (int i = tid; i < 180 * 48 * 2; i += 256) {
    int j = i & 1, t = i >> 1;
    int o = t % 48, kp = t / 48;
    int k = 2 * kp + j;
    wnsp[i] = (o < 40) ? w3[o * 360 + k] / wnorm[o] : 0.f;
  }
  if (tid < 128) {
    int img = tid >> 4, m = tid & 15;
    int base = (m >> 2) * 6 + (m & 3);
    float s = 0.f;
    for (int k = 0; k < 360; k++) { float v = xs[img][base + koff[k]]; s += v * v; }
    xn[img][m] = __builtin_sqrtf(s + EPSC) + qv;
  }
  __syncthreads();

  const int wave = tid >> 5, lane = tid & 31;
  const int lm = lane & 15, lh = lane >> 4;
  const int img = wave;
  const int abase = (lm >> 2) * 6 + (lm & 3);

  for (int nt = 0; nt < 3; nt++) {
    v8f acc = {0.f, 0.f, 0.f, 0.f, 0.f, 0.f, 0.f, 0.f};
    for (int ks = 0; ks < 90; ks++) {
      int k0 = ks * 4 + lh * 2;
      int kp = ks * 2 + lh;
      v2f a;
      a.x = xs[img][abase + koff[k0]];
      a.y = xs[img][abase + koff[k0 + 1]];
      v2f b = *(const v2f*)&wnsp[(kp * 48 + nt * 16 + lm) * 2]; // ds_load_b64
      acc = __builtin_amdgcn_wmma_f32_16x16x4_f32(false, a, false, b,
                                                  (short)0, acc, false, false);
    }
    int o = nt * 16 + lm;
    float ex = expo[o];
    float r = scs_act(acc[0], xn[img][8 * lh + 0], ex);
    r = maxabs2(r, scs_act(acc[1], xn[img][8 * lh + 1], ex));
    r = maxabs2(r, scs_act(acc[2], xn[img][8 * lh + 2], ex));
    r = maxabs2(r, scs_act(acc[3], xn[img][8 * lh + 3], ex));
    r = maxabs2(r, scs_act(acc[4], xn[img][8 * lh + 4], ex));
    r = maxabs2(r, scs_act(acc[5], xn[img][8 * lh + 5], ex));
    r = maxabs2(r, scs_act(acc[6], xn[img][8 * lh + 6], ex));
    r = maxabs2(r, scs_act(acc[7], xn[img][8 * lh + 7], ex));
    r = maxabs2(r, __shfl_xor(r, 16, 32));     // combine M halves (same o)
    if (lane < 16) po[img][o] = r;
  }
  // final linear, lanes 0..9 each produce one class (LDS in-order per wave)
  if (lane < 10) {
    float s = bo[lane];
    for (int o = 0; o < 40; o++) s += po[img][o] * wo[lane * 40 + o];
    out[(size_t)(n0 + img) * 10 + lane] = s;
  }
}

// ---------------------------------------------------------------------------
extern "C" void kernel_launch(void* const* d_in, const int* in_sizes, int n_in,
                              void* d_out, int out_size, void* d_ws, size_t ws_size,
                              hipStream_t stream) {
  const float* x   = (const float*)d_in[0];
  const float* w1  = (const float*)d_in[1];
  const float* p1  = (const float*)d_in[2];
  const float* q1  = (const float*)d_in[3];
  const float* w2d = (const float*)d_in[4];
  const float* p2d = (const float*)d_in[5];
  const float* q2d = (const float*)d_in[6];
  const float* w2p = (const float*)d_in[7];
  const float* p2p = (const float*)d_in[8];
  const float* q2p = (const float*)d_in[9];
  const float* w3  = (const float*)d_in[10];
  const float* p3  = (const float*)d_in[11];
  const float* q3  = (const float*)d_in[12];
  const float* wo  = (const float*)d_in[13];
  const float* bo  = (const float*)d_in[14];
  float* out = (float*)d_out;

  float* ws = (float*)d_ws;
  float* t1p  = ws;                               // (1024,80,14,14)  64.2 MB
  float* t2d  = t1p + (size_t)1024 * 80 * 196;    // (1024,80,12,12)  47.2 MB
  float* t2pp = t2d + (size_t)1024 * 80 * 144;    // (1024,40,6,6)     5.9 MB

  k_scs1<<<1024, 256, 0, stream>>>(x, w1, p1, q1, t1p);
  k_dw<<<46080, 256, 0, stream>>>(t1p, w2d, p2d, q2d, t2d);
  k_pw<<<1024, 256, 0, stream>>>(t2d, w2p, p2p, q2p, t2pp);
  k_c3<<<128, 256, 0, stream>>>(t2pp, w3, p3, q3, wo, bo, out);
}